// MacheteLinearCrossEntropy_75977971466920
// MI455X (gfx1250) — compile-verified
//
#include <hip/hip_runtime.h>
#include <math.h>

typedef float v2f __attribute__((ext_vector_type(2)));
typedef float v8f __attribute__((ext_vector_type(8)));

#define IGNORE_INDEX (-100)

// Problem shape from reference: x (N,H) fp32, weight (V,H) fp32, target (N,) int
constexpr int N = 8192;
constexpr int H = 4096;
constexpr int V = 32000;

constexpr int BN     = 128;             // rows per workgroup
constexpr int BV     = 64;              // cols per sub-tile
constexpr int SUBT   = 4;               // sub-tiles per WG -> 256 cols per WG
constexpr int VB     = V / (BV * SUBT); // 125 v-blocks
constexpr int KC     = 32;              // K chunk staged in LDS
constexpr int LDA    = KC + 4;          // 36: conflict-free stride for b64 frag reads
constexpr int NCHUNK = H / KC;          // 128

// ---------------------------------------------------------------------------
// Async global->LDS staging of one K-chunk (A: 128xKC, B: 64xKC), tracked by
// ASYNCcnt (gfx1250 GLOBAL_LOAD_ASYNC_TO_LDS_B128). Per-thread: 4 A + 2 B tiles.
// ---------------------------------------------------------------------------
__device__ __forceinline__ void async_stage(const float* __restrict__ xp,
                                            const float* __restrict__ wp,
                                            float* As, float* Bs, int tid)
{
  #pragma unroll
  for (int it = 0; it < 4; ++it) {
    int idx = it * 256 + tid, r = idx >> 3, c4 = idx & 7;
    const float* g = xp + (size_t)r * H + c4 * 4;
    unsigned l = (unsigned)(uintptr_t)(As + r * LDA + c4 * 4);
    asm volatile("global_load_async_to_lds_b128 %0, %1, off" :: "v"(l), "v"(g) : "memory");
  }
  #pragma unroll
  for (int it = 0; it < 2; ++it) {
    int idx = it * 256 + tid, r = idx >> 3, c4 = idx & 7;
    const float* g = wp + (size_t)r * H + c4 * 4;
    unsigned l = (unsigned)(uintptr_t)(Bs + r * LDA + c4 * 4);
    asm volatile("global_load_async_to_lds_b128 %0, %1, off" :: "v"(l), "v"(g) : "memory");
  }
}

__device__ __forceinline__ void wait_async_all() {
  asm volatile("s_wait_asynccnt 0x0" ::: "memory");
}

// ---------------------------------------------------------------------------
// K1: fused GEMM tile + online softmax partials.
// ws_max/ws_sum: [VB][N] (v-block major, coalesced for K2). ws_tgt: [N].
// ---------------------------------------------------------------------------
__global__ __launch_bounds__(256) void k1_lse_partial(
    const float* __restrict__ x, const float* __restrict__ w,
    const int* __restrict__ target,
    float* __restrict__ ws_max, float* __restrict__ ws_sum,
    float* __restrict__ ws_tgt)
{
  __shared__ float Astage[2][BN * LDA];   // 2 x 128 x 36 floats (36,864 B)
  __shared__ float Bstage[2][BV * LDA];   // 2 x  64 x 36 floats (18,432 B)
  __shared__ float partMax[BN * 2];       // per-row partials from the 2 col-halves
  __shared__ float partSum[BN * 2];
  __shared__ float runMax[BN];
  __shared__ float runSum[BN];
  __shared__ int   tgts[BN];

  const int tid   = threadIdx.x;
  const int lane  = tid & 31;
  const int wid   = tid >> 5;
  const int waveM = wid >> 1;      // 0..3 -> 32-row slice
  const int waveN = wid & 1;       // 0..1 -> 32-col slice
  const int lmod  = lane & 15;     // M/N index within 16
  const int lhalf = lane >> 4;     // K pair / row-half select

  const int rowBase = blockIdx.x * BN;
  const float* xrow = x + (size_t)rowBase * H;

  if (tid < BN) {
    runMax[tid] = -INFINITY;
    runSum[tid] = 0.f;
    tgts[tid]   = target[rowBase + tid];
  }
  __syncthreads();

  for (int st = 0; st < SUBT; ++st) {
    const int colBase = (blockIdx.y * SUBT + st) * BV;
    const float* wcol = w + (size_t)colBase * H;

    v8f zero = {0.f, 0.f, 0.f, 0.f, 0.f, 0.f, 0.f, 0.f};
    v8f acc[2][2];
    acc[0][0] = zero; acc[0][1] = zero; acc[1][0] = zero; acc[1][1] = zero;

    // Prologue: stage chunk 0 into buffer 0.
    async_stage(xrow, wcol, Astage[0], Bstage[0], tid);

    for (int kc = 0; kc < NCHUNK; ++kc) {
      float* Ac = Astage[kc & 1];
      float* Bc = Bstage[kc & 1];

      wait_async_all();      // own async loads for chunk kc have landed in LDS
      __syncthreads();       // everyone's landed; prev compute done -> other buf free

      if (kc + 1 < NCHUNK)
        async_stage(xrow + (kc + 1) * KC, wcol + (kc + 1) * KC,
                    Astage[(kc + 1) & 1], Bstage[(kc + 1) & 1], tid);

      #pragma unroll
      for (int k4 = 0; k4 < KC / 4; ++k4) {
        const int kk = k4 * 4 + lhalf * 2;   // K = 4*k4 + 2*(lane/16) + {0,1}
        v2f a0 = *(const v2f*)(Ac + (waveM * 32 + 0 * 16 + lmod) * LDA + kk);
        v2f a1 = *(const v2f*)(Ac + (waveM * 32 + 1 * 16 + lmod) * LDA + kk);
        v2f b0 = *(const v2f*)(Bc + (waveN * 32 + 0 * 16 + lmod) * LDA + kk);
        v2f b1 = *(const v2f*)(Bc + (waveN * 32 + 1 * 16 + lmod) * LDA + kk);
        acc[0][0] = __builtin_amdgcn_wmma_f32_16x16x4_f32(false, a0, false, b0, (short)0, acc[0][0], false, false);
        acc[0][1] = __builtin_amdgcn_wmma_f32_16x16x4_f32(false, a0, false, b1, (short)0, acc[0][1], false, false);
        acc[1][0] = __builtin_amdgcn_wmma_f32_16x16x4_f32(false, a1, false, b0, (short)0, acc[1][0], false, false);
        acc[1][1] = __builtin_amdgcn_wmma_f32_16x16x4_f32(false, a1, false, b1, (short)0, acc[1][1], false, false);
      }
    }

    // ---- Online (max, sum-exp) update straight from accumulators. ----
    // C/D layout: VGPR i -> row = waveM*32 + mt*16 + i + 8*(lane/16), col = waveN*32 + nt*16 + lane%16.
    #pragma unroll
    for (int mt = 0; mt < 2; ++mt) {
      #pragma unroll
      for (int i = 0; i < 8; ++i) {
        float v0 = acc[mt][0][i];
        float v1 = acc[mt][1][i];
        const int row = waveM * 32 + mt * 16 + i + 8 * lhalf;

        // target-logit extraction from registers
        const int t  = tgts[row];
        const int c0 = colBase + waveN * 32 + lmod;
        if (t == c0)      ws_tgt[rowBase + row] = v0;
        if (t == c0 + 16) ws_tgt[rowBase + row] = v1;

        // reduce across the 16 lanes holding this row (masks < 16 stay in-half)
        float m = fmaxf(v0, v1);
        m = fmaxf(m, __shfl_xor(m, 1));
        m = fmaxf(m, __shfl_xor(m, 2));
        m = fmaxf(m, __shfl_xor(m, 4));
        m = fmaxf(m, __shfl_xor(m, 8));
        float s = __expf(v0 - m) + __expf(v1 - m);
        s += __shfl_xor(s, 1);
        s += __shfl_xor(s, 2);
        s += __shfl_xor(s, 4);
        s += __shfl_xor(s, 8);
        if (lmod == 0) {
          partMax[row * 2 + waveN] = m;
          partSum[row * 2 + waveN] = s;
        }
      }
    }
    __syncthreads();
    if (tid < BN) {
      float m0 = partMax[tid * 2 + 0], m1 = partMax[tid * 2 + 1];
      float m2 = fmaxf(m0, m1);
      float s2 = partSum[tid * 2 + 0] * __expf(m0 - m2)
               + partSum[tid * 2 + 1] * __expf(m1 - m2);
      float om = runMax[tid], nm = fmaxf(om, m2);
      runSum[tid] = runSum[tid] * __expf(om - nm) + s2 * __expf(m2 - nm);
      runMax[tid] = nm;
    }
    __syncthreads();
  }

  if (tid < BN) {
    ws_max[(size_t)blockIdx.y * N + rowBase + tid] = runMax[tid];
    ws_sum[(size_t)blockIdx.y * N + rowBase + tid] = runSum[tid];
  }
}

// ---------------------------------------------------------------------------
// K2: combine VB partials per row -> per-row NLL.
// ---------------------------------------------------------------------------
__global__ __launch_bounds__(256) void k2_rows(
    const int* __restrict__ target,
    const float* __restrict__ ws_max, const float* __restrict__ ws_sum,
    const float* __restrict__ ws_tgt, float* __restrict__ ws_nll)
{
  const int n = blockIdx.x * 256 + threadIdx.x;
  float M = -INFINITY;
  for (int i = 0; i < VB; ++i) M = fmaxf(M, ws_max[(size_t)i * N + n]);
  float S = 0.f;
  for (int i = 0; i < VB; ++i)
    S += ws_sum[(size_t)i * N + n] * __expf(ws_max[(size_t)i * N + n] - M);
  float lse = M + logf(S);
  int t = target[n];
  ws_nll[n] = (t != IGNORE_INDEX) ? (lse - ws_tgt[n]) : 0.f;
}

// ---------------------------------------------------------------------------
// K3: mean over valid rows -> scalar.
// ---------------------------------------------------------------------------
__global__ __launch_bounds__(256) void k3_reduce(
    const int* __restrict__ target, const float* __restrict__ ws_nll,
    float* __restrict__ out)
{
  __shared__ float ss[256];
  __shared__ int   sc[256];
  const int tid = threadIdx.x;
  float s = 0.f; int c = 0;
  for (int i = tid; i < N; i += 256) {
    if (target[i] != IGNORE_INDEX) { s += ws_nll[i]; c++; }
  }
  ss[tid] = s; sc[tid] = c;
  __syncthreads();
  for (int o = 128; o > 0; o >>= 1) {
    if (tid < o) { ss[tid] += ss[tid + o]; sc[tid] += sc[tid + o]; }
    __syncthreads();
  }
  if (tid == 0) out[0] = ss[0] / (float)(sc[0] > 0 ? sc[0] : 1);
}

// ---------------------------------------------------------------------------
extern "C" void kernel_launch(void* const* d_in, const int* in_sizes, int n_in,
                              void* d_out, int out_size, void* d_ws, size_t ws_size,
                              hipStream_t stream) {
  const float* x   = (const float*)d_in[0];
  const float* w   = (const float*)d_in[1];
  const int*   tgt = (const int*)d_in[2];

  float* ws      = (float*)d_ws;
  float* ws_tgt  = ws;                         // N floats
  float* ws_max  = ws_tgt + N;                 // VB*N floats
  float* ws_sum  = ws_max + (size_t)VB * N;    // VB*N floats
  float* ws_nll  = ws_sum + (size_t)VB * N;    // N floats  (total ~8.3 MB)

  dim3 g1(N / BN, VB);   // x = n-blocks fastest -> x stays L2-resident, W streams once
  k1_lse_partial<<<g1, 256, 0, stream>>>(x, w, tgt, ws_max, ws_sum, ws_tgt);
  k2_rows<<<N / 256, 256, 0, stream>>>(tgt, ws_max, ws_sum, ws_tgt, ws_nll);
  k3_reduce<<<1, 256, 0, stream>>>(tgt, ws_nll, (float*)d_out);
}